// GCN_20942260536007
// MI455X (gfx1250) — compile-verified
//
#include <hip/hip_runtime.h>
#include <hip/hip_bf16.h>

// ---- problem constants (match reference) ----
#define NN   100000
#define EE   1600000
#define GG   128
#define HH   128
#define CC   10
#define EPSV 1e-5f

typedef __attribute__((ext_vector_type(2))) float v2f;
typedef __attribute__((ext_vector_type(8))) float v8f;
typedef __attribute__((ext_vector_type(4))) unsigned int v4u;
typedef __attribute__((ext_vector_type(4))) int v4i;
typedef __attribute__((ext_vector_type(8))) int v8i;

// ---------------- fill ----------------
__global__ void fill_f32(float* __restrict__ p, float v, long n) {
    long i = (long)blockIdx.x * blockDim.x + threadIdx.x;
    if (i < n) p[i] = v;
}

// ---------------- degree count over edges (self-loop pre-seeded as 1.0) ----------------
__global__ void deg_count(const int* __restrict__ ei, float* __restrict__ deg) {
    int e = blockIdx.x * blockDim.x + threadIdx.x;
    if (e < EE) atomicAdd(&deg[ei[EE + e]], 1.0f);
}

// ---------------- deg -> dis = rsqrt(deg), in place ----------------
__global__ void deg_to_dis(float* __restrict__ d) {
    int i = blockIdx.x * blockDim.x + threadIdx.x;
    if (i < NN) d[i] = rsqrtf(d[i]);   // deg >= 1 always (self loop)
}

// ---------------- GEMM: Out[N,128] = A[N,128] @ W[128,128] via v_wmma_f32_16x16x4_f32 ----
// block = 256 threads = 8 waves; wave w owns 16 rows; block panel = 128 rows.
// W staged in LDS pair-interleaved + XOR-swizzled so each B frag is one ds_load_b64.
// A panel staged by the Tensor Data Mover with 2-DWORD row padding (stride 130 floats).
#define AROW 130
__global__ void __launch_bounds__(256)
gemm_wmma_f32(const float* __restrict__ A, const float* __restrict__ W,
              float* __restrict__ Out, int nrows) {
    __shared__ float sW[64 * 256];       // 64 k-pairs x 128 cols x 2  (64 KB)
    __shared__ float sA[128 * AROW];     // 128 rows x (128+2 pad)     (65 KB)

    const int tid = threadIdx.x;

    // --- stage W: sW[(k>>1)*256 + ((col*2) ^ (((k>>1)&1)<<5)) + (k&1)] = W[k][col]
    for (int i = tid; i < 128 * 128; i += 256) {
        const int k = i >> 7, col = i & 127;
        const int pr = k >> 1;
        sW[pr * 256 + (((col * 2) ^ ((pr & 1) << 5))) + (k & 1)] = W[i];
    }

    // --- stage A panel via TDM (wave 0 issues; DMA ignores EXEC, gate keeps it single)
    if (tid < 32) {
        const unsigned ldsOff = (unsigned)(unsigned long long)(const void*)sA;
        const unsigned long long ga =
            (unsigned long long)(const void*)(A + (size_t)blockIdx.x * 128 * 128);
        int remRows = nrows - blockIdx.x * 128;
        if (remRows > 128) remRows = 128;
        const unsigned dim0 = (unsigned)remRows * 128u;  // OOB reads -> zero fill

        v4u g0;
        g0.x = 1u;                                  // count=1, user mode, no gather
        g0.y = ldsOff;                              // lds_addr
        g0.z = (unsigned)ga;                        // global_addr[31:0]
        g0.w = (unsigned)((ga >> 32) & 0x1FFFFFFull) | 0x80000000u;  // addr[56:32] | type=2

        v8i g1;
        g1[0] = (int)((2u << 16)      // data_size = 4B
                    | (1u << 20)      // pad_enable
                    | (6u << 22)      // pad_interval: every 128 DWORDs
                    | (1u << 25));    // pad_amount: 2 DWORDs  -> row stride 130
        g1[1] = (int)((dim0 & 0xFFFFu) << 16);                  // tensor_dim0 lo16
        g1[2] = (int)(((dim0 >> 16) & 0xFFFFu) | (1u << 16));   // tensor_dim0 hi | tensor_dim1=1
        g1[3] = (int)(16384u << 16);                            // tile_dim0 = 16384
        g1[4] = 1;                                              // tile_dim1 = 1
        g1[5] = 16384;                                          // tensor_dim0_stride
        g1[6] = 0;
        g1[7] = 0;

        v4i gz = {0, 0, 0, 0};
#if __clang_major__ >= 23
        v8i gz8 = {0, 0, 0, 0, 0, 0, 0, 0};
        __builtin_amdgcn_tensor_load_to_lds(g0, g1, gz, gz, gz8, 0);
#else
        __builtin_amdgcn_tensor_load_to_lds(g0, g1, gz, gz, 0);
#endif
        __builtin_amdgcn_s_wait_tensorcnt(0);
    }
    __syncthreads();

    const int wave = tid >> 5;
    const int lane = tid & 31;
    const int half = lane >> 4;   // 0 or 1
    const int l16  = lane & 15;
    const int rowBase = (blockIdx.x * 8 + wave) * 16;
    if (rowBase >= nrows) return;  // wave-uniform: EXEC all-1s inside

    const int rowLocal = wave * 16 + l16;
    v8f acc[8];
    #pragma unroll
    for (int ct = 0; ct < 8; ++ct) { v8f z = {}; acc[ct] = z; }

    // A frag (ISA 7.12.2, 32-bit A 16x4): lane holds {A[row][krow], A[row][krow+1]},
    // krow = kb*4 + half*2  -> contiguous aligned float2 in padded sA.
    const float* aptr = sA + rowLocal * AROW + half * 2;

    #pragma unroll 4
    for (int kb = 0; kb < 32; ++kb) {
        const v2f a = *(const v2f*)(aptr + kb * 4);
        const int pr = kb * 2 + half;              // k-pair row
        const float* bbase = sW + pr * 256;
        const int sw = (pr & 1) << 5;
        #pragma unroll
        for (int ct = 0; ct < 8; ++ct) {
            const int col = ct * 16 + l16;
            const v2f b = *(const v2f*)(bbase + ((col * 2) ^ sw));
            acc[ct] = __builtin_amdgcn_wmma_f32_16x16x4_f32(
                false, a, false, b, (short)0, acc[ct], false, false);
        }
    }

    // C/D layout: VGPR i -> M = i (lanes 0-15) / i+8 (lanes 16-31), N = l16
    #pragma unroll
    for (int ct = 0; ct < 8; ++ct)
        #pragma unroll
        for (int i = 0; i < 8; ++i)
            Out[(size_t)(rowBase + half * 8 + i) * 128 + ct * 16 + l16] = acc[ct][i];
}

// ---------------- edge scatter: agg[dst] += h[src] * dis[src]*dis[dst] ----------------
__global__ void edge_scatter(const int* __restrict__ ei, const float* __restrict__ dis,
                             const float* __restrict__ h, float* __restrict__ agg) {
    const int lane = threadIdx.x & 31;
    int wid = (blockIdx.x * blockDim.x + threadIdx.x) >> 5;
    const int nw = (gridDim.x * blockDim.x) >> 5;
    for (int e = wid; e < EE; e += nw) {
        const int s = ei[e];
        const int d = ei[EE + e];
        const float w = dis[s] * dis[d];
        const float4 v = *(const float4*)(h + (size_t)s * 128 + lane * 4);
        float* o = agg + (size_t)d * 128 + lane * 4;
        atomicAdd(o + 0, v.x * w);
        atomicAdd(o + 1, v.y * w);
        atomicAdd(o + 2, v.z * w);
        atomicAdd(o + 3, v.w * w);
    }
}

// ---------------- BN pre-pass: add self-loop + bias in place, accumulate sum/sumsq ------
__global__ void __launch_bounds__(256)
bn_pre(float* __restrict__ agg, const float* __restrict__ hlin,
       const float* __restrict__ dis, const float* __restrict__ bias,
       float* __restrict__ sum, float* __restrict__ sumsq) {
    __shared__ float sS[256], sQ[256];
    const int tid = threadIdx.x;
    const int c  = tid & 127;
    const int rg = tid >> 7;
    const float bc = bias[c];
    float s = 0.f, q = 0.f;
    const int rowBase = blockIdx.x * 64;
    for (int r = rowBase + rg; r < rowBase + 64; r += 2) {
        if (r < NN) {
            const size_t idx = (size_t)r * 128 + c;
            const float di = dis[r];
            const float val = agg[idx] + di * di * hlin[idx] + bc;
            agg[idx] = val;
            s += val;
            q += val * val;
        }
    }
    sS[tid] = s; sQ[tid] = q;
    __syncthreads();
    if (tid < 128) {
        atomicAdd(&sum[c],   sS[tid] + sS[tid + 128]);
        atomicAdd(&sumsq[c], sQ[tid] + sQ[tid + 128]);
    }
}

// ---------------- BN stats -> scale/shift ----------------
__global__ void bn_stats(const float* __restrict__ sum, const float* __restrict__ sumsq,
                         const float* __restrict__ g, const float* __restrict__ be,
                         float* __restrict__ scale, float* __restrict__ shift) {
    const int c = threadIdx.x;  // 128 threads
    const float inv = 1.0f / (float)NN;
    const float mu  = sum[c] * inv;
    const float var = sumsq[c] * inv - mu * mu;
    const float sc  = g[c] * rsqrtf(var + EPSV);
    scale[c] = sc;
    shift[c] = be[c] - mu * sc;
}

// ---------------- normalize + ReLU ----------------
__global__ void bn_norm_relu(const float* __restrict__ agg, const float* __restrict__ scale,
                             const float* __restrict__ shift, float* __restrict__ out) {
    const long i = (long)blockIdx.x * blockDim.x + threadIdx.x;
    if (i < (long)NN * 128) {
        const int c = (int)(i & 127);
        out[i] = fmaxf(fmaf(agg[i], scale[c], shift[c]), 0.0f);
    }
}

// ---------------- global add pool ----------------
__global__ void pool_scatter(const float* __restrict__ h, const int* __restrict__ batch,
                             float* __restrict__ pooled) {
    const long i = (long)blockIdx.x * blockDim.x + threadIdx.x;
    if (i < (long)NN * 128) {
        const int node = (int)(i >> 7);
        const int c    = (int)(i & 127);
        atomicAdd(&pooled[(size_t)batch[node] * 128 + c], h[i]);
    }
}

// ---------------- classifier: out[G,C] = pooled @ Wc + bc ----------------
__global__ void classifier(const float* __restrict__ pooled, const float* __restrict__ Wc,
                           const float* __restrict__ bc, float* __restrict__ out) {
    const int idx = blockIdx.x * blockDim.x + threadIdx.x;
    if (idx < GG * CC) {
        const int g = idx / CC;
        const int c = idx % CC;
        float acc = bc[c];
        #pragma unroll 8
        for (int k = 0; k < 128; ++k)
            acc = fmaf(pooled[g * 128 + k], Wc[k * CC + c], acc);
        out[idx] = acc;
    }
}

extern "C" void kernel_launch(void* const* d_in, const int* in_sizes, int n_in,
                              void* d_out, int out_size, void* d_ws, size_t ws_size,
                              hipStream_t stream) {
    // inputs (setup_inputs dict order)
    const float* x     = (const float*)d_in[0];
    const int*   ei    = (const int*)d_in[1];   // edge_index [2,E] flat
    const int*   batch = (const int*)d_in[2];
    const float* W[3]  = { (const float*)d_in[3], (const float*)d_in[7],  (const float*)d_in[11] };
    const float* bL[3] = { (const float*)d_in[4], (const float*)d_in[8],  (const float*)d_in[12] };
    const float* gL[3] = { (const float*)d_in[5], (const float*)d_in[9],  (const float*)d_in[13] };
    const float* beL[3]= { (const float*)d_in[6], (const float*)d_in[10], (const float*)d_in[14] };
    const float* Wc    = (const float*)d_in[15];
    const float* bc    = (const float*)d_in[16];
    float* out = (float*)d_out;

    // workspace carve-up (floats): ~155 MB total
    float* ws = (float*)d_ws;
    const size_t NH = (size_t)NN * 128;
    float* dis    = ws;                 // N
    float* hlin   = dis + NN;           // N*128
    float* agg    = hlin + NH;          // N*128
    float* hcur   = agg + NH;           // N*128
    float* sums   = hcur + NH;          // 128
    float* sumsq  = sums + 128;         // 128
    float* scale  = sumsq + 128;        // 128
    float* shift  = scale + 128;        // 128
    float* pooled = shift + 128;        // G*128

    const int T = 256;
    const int gridElem = (int)((NH + T - 1) / T);              // 50000
    const int gridGemm = (NN / 16 + 7) / 8;                    // 782
    const int gridEdge = 4096;                                 // wave-per-edge grid-stride
    const int gridBn   = (NN + 63) / 64;                       // 1563

    // degree / dis (self-loop contributes 1.0 seed)
    fill_f32<<<(NN + T - 1) / T, T, 0, stream>>>(dis, 1.0f, NN);
    deg_count<<<(EE + T - 1) / T, T, 0, stream>>>(ei, dis);
    deg_to_dis<<<(NN + T - 1) / T, T, 0, stream>>>(dis);

    const float* xin = x;
    for (int l = 0; l < 3; ++l) {
        gemm_wmma_f32<<<gridGemm, T, 0, stream>>>(xin, W[l], hlin, NN);
        fill_f32<<<gridElem, T, 0, stream>>>(agg, 0.0f, (long)NH);
        edge_scatter<<<gridEdge, T, 0, stream>>>(ei, dis, hlin, agg);
        fill_f32<<<1, T, 0, stream>>>(sums, 0.0f, 256);        // sums+sumsq contiguous
        bn_pre<<<gridBn, T, 0, stream>>>(agg, hlin, dis, bL[l], sums, sumsq);
        bn_stats<<<1, 128, 0, stream>>>(sums, sumsq, gL[l], beL[l], scale, shift);
        bn_norm_relu<<<gridElem, T, 0, stream>>>(agg, scale, shift, hcur);
        xin = hcur;
    }

    fill_f32<<<(GG * 128 + T - 1) / T, T, 0, stream>>>(pooled, 0.0f, GG * 128);
    pool_scatter<<<gridElem, T, 0, stream>>>(hcur, batch, pooled);
    classifier<<<(GG * CC + T - 1) / T, T, 0, stream>>>(pooled, Wc, bc, out);
}